// LINK_21646635172435
// MI455X (gfx1250) — compile-verified
//
#include <hip/hip_runtime.h>
#include <hip/hip_bf16.h>
#include <limits.h>

// ---------------------------------------------------------------------------
// GCN link prediction forward:  out[i,c] = b[c] + sum_{e: row[e]==i} W[c, col[e]]
// Strategy: build CSR (int atomics only), transpose W once (async LDS tiling),
// then one wave32 per node does a register-resident segment reduction.
// ---------------------------------------------------------------------------

#ifndef LINK_WAVE
#define LINK_WAVE 32
#endif

#if defined(__has_builtin)
#if __has_builtin(__builtin_amdgcn_global_load_async_to_lds_b128)
#define LINK_HAVE_ASYNC_LDS 1
#endif
#if __has_builtin(__builtin_amdgcn_s_wait_asynccnt)
#define LINK_HAVE_WAIT_ASYNC_BUILTIN 1
#endif
#endif

#if defined(LINK_HAVE_ASYNC_LDS)
typedef int link_v4i __attribute__((__vector_size__(16)));
typedef link_v4i __attribute__((address_space(1))) link_v4i_glb;
typedef link_v4i __attribute__((address_space(3))) link_v4i_lds;
#endif

__device__ __forceinline__ void link_wait_asynccnt0() {
#if defined(LINK_HAVE_ASYNC_LDS)
#if defined(LINK_HAVE_WAIT_ASYNC_BUILTIN)
  __builtin_amdgcn_s_wait_asynccnt(0);
#else
  asm volatile("s_wait_asynccnt 0" ::: "memory");
#endif
#endif
}

// ---------------- K0: init counts + min sentinel ---------------------------
__global__ void link_init_kernel(int* __restrict__ cnt, int N, int* __restrict__ minRow) {
  int i = blockIdx.x * blockDim.x + threadIdx.x;
  if (i < N) cnt[i] = 0;
  if (i == 0) *minRow = INT_MAX;
}

// ---------------- K1: min over edge rows (mirrors row - row.min()) ---------
__global__ void link_min_kernel(const int* __restrict__ row, int E, int* __restrict__ minRow) {
  __shared__ int smem[256];
  int v = INT_MAX;
  for (int i = blockIdx.x * blockDim.x + threadIdx.x; i < E; i += gridDim.x * blockDim.x)
    v = min(v, row[i]);
  smem[threadIdx.x] = v;
  __syncthreads();
  for (int o = 128; o > 0; o >>= 1) {
    if (threadIdx.x < o) smem[threadIdx.x] = min(smem[threadIdx.x], smem[threadIdx.x + o]);
    __syncthreads();
  }
  if (threadIdx.x == 0) atomicMin(minRow, smem[0]);
}

// ---------------- K2: per-node edge counts ---------------------------------
__global__ void link_count_kernel(const int* __restrict__ row, int E,
                                  const int* __restrict__ minRow, int* __restrict__ cnt) {
  int m = *minRow;
  int i = blockIdx.x * blockDim.x + threadIdx.x;
  if (i < E) atomicAdd(&cnt[row[i] - m], 1);
}

// ---------------- K3: exclusive scan (single workgroup, 1024 thr) ----------
__global__ void link_scan_kernel(const int* __restrict__ cnt, int N,
                                 int* __restrict__ start, int* __restrict__ cursor) {
  __shared__ int buf[1024];
  __shared__ int carry;
  const int tid = threadIdx.x;
  if (tid == 0) carry = 0;
  __syncthreads();
  for (int base = 0; base < N; base += 1024) {
    int v = (base + tid < N) ? cnt[base + tid] : 0;
    buf[tid] = v;
    __syncthreads();
    for (int off = 1; off < 1024; off <<= 1) {
      int t = (tid >= off) ? buf[tid - off] : 0;
      __syncthreads();
      buf[tid] += t;
      __syncthreads();
    }
    int incl = buf[tid];
    int c = carry;
    int excl = c + incl - v;
    if (base + tid < N) { start[base + tid] = excl; cursor[base + tid] = excl; }
    __syncthreads();
    if (tid == 1023) carry = c + buf[1023];
    __syncthreads();
  }
}

// ---------------- K4: scatter cols into CSR order --------------------------
__global__ void link_scatter_kernel(const int* __restrict__ row, const int* __restrict__ col,
                                    int E, const int* __restrict__ minRow,
                                    int* __restrict__ cursor, int* __restrict__ sortedCol) {
  int m = *minRow;
  int i = blockIdx.x * blockDim.x + threadIdx.x;
  if (i < E) {
    int r = row[i] - m;
    int slot = atomicAdd(&cursor[r], 1);
    sortedCol[slot] = col[i];
  }
}

// ---------------- K5: transpose W[C,N] -> Wt[N,C] via async LDS tiles ------
// Tile: 64 classes x 64 nodes. Load phase streams global->LDS (async, no VGPR
// round trip), store phase emits 256B-coalesced rows of Wt.
__global__ void link_transpose_kernel(const float* __restrict__ W, int N,
                                      float* __restrict__ Wt) {
  __shared__ float tile[64][68];  // +4 float pad keeps 16B alignment, limits bank conflicts
  const int t = threadIdx.x;      // 256 threads
  const int j0 = blockIdx.x * 64;

  // Load: 4 passes x (16 class-rows x 64 node-cols per pass), 16B per thread.
  for (int pass = 0; pass < 4; ++pass) {
    int c  = pass * 16 + (t >> 4);
    int jj = (t & 15) * 4;
    int j  = j0 + jj;
    if (j < N) {  // N % 4 == 0 so the 16B chunk is fully in-bounds
#if defined(LINK_HAVE_ASYNC_LDS)
      link_v4i_glb* g = (link_v4i_glb*)(W + (size_t)c * N + j);
      link_v4i_lds* l = (link_v4i_lds*)&tile[c][jj];
      __builtin_amdgcn_global_load_async_to_lds_b128(g, l, 0, 0);
#else
      const float4 v = *reinterpret_cast<const float4*>(W + (size_t)c * N + j);
      *reinterpret_cast<float4*>(&tile[c][jj]) = v;
#endif
    }
  }
  link_wait_asynccnt0();
  __syncthreads();

  // Store: 16 passes, each 64-thread group writes one 256B row of Wt.
  for (int pass = 0; pass < 16; ++pass) {
    int j = pass * 4 + (t >> 6);
    int c = t & 63;
    if (j0 + j < N) Wt[(size_t)(j0 + j) * 64 + c] = tile[c][j];
  }
}

// ---------------- K6: one wave per node: gather + register reduce ----------
__global__ void link_aggregate_kernel(const float* __restrict__ Wt, const float* __restrict__ b,
                                      const int* __restrict__ start, const int* __restrict__ cnt,
                                      const int* __restrict__ sortedCol, int N,
                                      float* __restrict__ out) {
  const int wid  = (int)((blockIdx.x * blockDim.x + threadIdx.x) >> 5);
  const int lane = threadIdx.x & (LINK_WAVE - 1);
  if (wid >= N) return;

  const int s = start[wid];
  const int n = cnt[wid];

  float accx = b[2 * lane];
  float accy = b[2 * lane + 1];

  for (int base = 0; base < n; base += LINK_WAVE) {
    int m = n - base;
    if (m > LINK_WAVE) m = LINK_WAVE;
    int myc = 0;
    if (lane < m) myc = sortedCol[s + base + lane];  // one coalesced b32 per 32 edges
    for (int k = 0; k < m; ++k) {
      int c = __shfl(myc, k, LINK_WAVE);             // ds_bpermute broadcast
      if (k + 6 < m) {
        int cp = __shfl(myc, k + 6, LINK_WAVE);
        __builtin_prefetch(Wt + (size_t)cp * 64, 0, 0);  // global_prefetch_b8
      }
      const float2 v = reinterpret_cast<const float2*>(Wt + (size_t)c * 64)[lane];
      accx += v.x;
      accy += v.y;
    }
  }

  // Stream out non-temporally: keep the 192MB L2 holding Wt, not the output.
  float* o = out + (size_t)wid * 64 + 2 * lane;
  __builtin_nontemporal_store(accx, o);
  __builtin_nontemporal_store(accy, o + 1);
}

// ---------------------------------------------------------------------------
extern "C" void kernel_launch(void* const* d_in, const int* in_sizes, int n_in,
                              void* d_out, int out_size, void* d_ws, size_t ws_size,
                              hipStream_t stream) {
  // inputs: 0=x (unused), 1=edge_index [2,E] int, 2=W [C,N] f32, 3=b [C] f32
  const int*   edge = (const int*)d_in[1];
  const float* W    = (const float*)d_in[2];
  const float* b    = (const float*)d_in[3];

  const int E = in_sizes[1] / 2;
  const int C = in_sizes[3];          // 64
  const int N = in_sizes[2] / C;      // 100000

  const int* row = edge;
  const int* col = edge + E;

  char* ws = (char*)d_ws;
  size_t off = 0;
  int* minRow = (int*)(ws + off);     off += 16;
  int* cnt    = (int*)(ws + off);     off += (size_t)N * sizeof(int);
  int* start  = (int*)(ws + off);     off += (size_t)N * sizeof(int);
  int* cursor = (int*)(ws + off);     off += (size_t)N * sizeof(int);
  int* sCol   = (int*)(ws + off);     off += (size_t)E * sizeof(int);
  off = (off + 255) & ~(size_t)255;
  float* Wt   = (float*)(ws + off);   off += (size_t)N * C * sizeof(float);

  float* out = (float*)d_out;

  const int nBlkN = (N + 255) / 256;
  const int nBlkE = (E + 255) / 256;

  link_init_kernel<<<nBlkN, 256, 0, stream>>>(cnt, N, minRow);
  link_min_kernel<<<512, 256, 0, stream>>>(row, E, minRow);
  link_count_kernel<<<nBlkE, 256, 0, stream>>>(row, E, minRow, cnt);
  link_scan_kernel<<<1, 1024, 0, stream>>>(cnt, N, start, cursor);
  link_scatter_kernel<<<nBlkE, 256, 0, stream>>>(row, col, E, minRow, cursor, sCol);
  link_transpose_kernel<<<(N + 63) / 64, 256, 0, stream>>>(W, N, Wt);
  link_aggregate_kernel<<<(N + 7) / 8, 256, 0, stream>>>(Wt, b, start, cnt, sCol, N, out);
}